// MEC_61443802137225
// MI455X (gfx1250) — compile-verified
//
#include <hip/hip_runtime.h>

// MI455X / gfx1250, wave32. Matrix math via V_WMMA_F32_16X16X4_F32 (exact f32
// semantics w.r.t. the reference). Key algebra: tr(c^k) for c = v1 v2^T / lam
// (8192x8192) equals tr(M^k) for M = v2^T v1 / lam (256x256) by trace
// cyclicity, collapsing ~4.4 TFLOP of B x B Taylor powers into ~2 GFLOP.
// The 128MB h intermediate stays resident in the 192MB global L2.
#define B_DIM 8192
#define E_DIM 256
#define H_DIM 4096
#define BN_EPS 1e-5f

typedef __attribute__((ext_vector_type(2))) float v2f;
typedef __attribute__((ext_vector_type(8))) float v8f;

static __device__ __forceinline__ v8f wmma4(v2f a, v2f b, v8f c) {
  // D = A(16x4) * B(4x16) + C, all f32.
  return __builtin_amdgcn_wmma_f32_16x16x4_f32(false, a, false, b, (short)0, c,
                                               false, false);
}

// ---------------------------------------------------------------------------
// Tiled transpose: Out[C][R] = In[R][C]. R, C multiples of 32.
// Used once for W1 (E x H) and W2 (H x E) so WMMA B-fragments become
// per-lane-contiguous b64 loads in the GEMM inner loops.
// ---------------------------------------------------------------------------
__global__ __launch_bounds__(256) void transpose_kernel(
    const float* __restrict__ In, float* __restrict__ Out, int R, int C) {
  __shared__ float tile[32][33];
  const int lane = threadIdx.x & 31;
  const int grp = threadIdx.x >> 5;  // 8 groups x 4 rows
  const int c = blockIdx.x * 32 + lane;
  const int r0 = blockIdx.y * 32 + grp * 4;
#pragma unroll
  for (int i = 0; i < 4; ++i)
    tile[grp * 4 + i][lane] = In[(size_t)(r0 + i) * C + c];
  __syncthreads();
  const int ro = blockIdx.y * 32 + lane;
  const int co0 = blockIdx.x * 32 + grp * 4;
#pragma unroll
  for (int i = 0; i < 4; ++i)
    Out[(size_t)(co0 + i) * R + ro] = tile[lane][grp * 4 + i];
}

// ---------------------------------------------------------------------------
// h = z @ W1.  z:[B,E] row-major, W1T:[H,E] (transposed W1), h:[B,H].
// 256 thr = 8 waves arranged 2(M) x 4(N); wave tile 32x32 -> 4 accumulators,
// inner loop: 4 global_load_b64 : 4 v_wmma (1:1), 4 independent acc chains.
// ---------------------------------------------------------------------------
__global__ __launch_bounds__(256) void gemm1_kernel(
    const float* __restrict__ Z, const float* __restrict__ W1T,
    float* __restrict__ Hout) {
  const int lane = threadIdx.x & 31;
  const int wave = threadIdx.x >> 5;
  const int lm = lane & 15;
  const int lh = lane >> 4;
  const int row0 = blockIdx.y * 64 + (wave >> 2) * 32;
  const int col0 = blockIdx.x * 128 + (wave & 3) * 32;

  v8f acc00 = {}, acc01 = {}, acc10 = {}, acc11 = {};
  const float* za0 = Z + (size_t)(row0 + lm) * E_DIM + 2 * lh;
  const float* za1 = za0 + (size_t)16 * E_DIM;
  const float* wb0 = W1T + (size_t)(col0 + lm) * E_DIM + 2 * lh;
  const float* wb1 = wb0 + (size_t)16 * E_DIM;
  for (int k0 = 0; k0 < E_DIM; k0 += 4) {
    v2f a0 = *(const v2f*)(za0 + k0);
    v2f a1 = *(const v2f*)(za1 + k0);
    v2f b0 = *(const v2f*)(wb0 + k0);
    v2f b1 = *(const v2f*)(wb1 + k0);
    acc00 = wmma4(a0, b0, acc00);
    acc01 = wmma4(a0, b1, acc01);
    acc10 = wmma4(a1, b0, acc10);
    acc11 = wmma4(a1, b1, acc11);
  }
  float* hp0 = Hout + (size_t)(row0 + 8 * lh) * H_DIM + col0 + lm;
  float* hp1 = hp0 + (size_t)16 * H_DIM;
#pragma unroll
  for (int r = 0; r < 8; ++r) {
    hp0[(size_t)r * H_DIM] = acc00[r];
    hp0[(size_t)r * H_DIM + 16] = acc01[r];
    hp1[(size_t)r * H_DIM] = acc10[r];
    hp1[(size_t)r * H_DIM + 16] = acc11[r];
  }
}

// ---------------------------------------------------------------------------
// Per-column BN stats over h (axis 0, biased var) -> fused affine:
//   aS[c] = gamma[c]*rsqrt(var+eps);  bS[c] = beta[c] - mu*aS[c]
// h re-read comes from L2 (128MB < 192MB).
// ---------------------------------------------------------------------------
__global__ __launch_bounds__(256) void bn_stats_kernel(
    const float* __restrict__ Hin, const float* __restrict__ gamma,
    const float* __restrict__ beta, float* __restrict__ aS,
    float* __restrict__ bS) {
  __shared__ float s_sum[256];
  __shared__ float s_sq[256];
  const int col = blockIdx.x * 64 + (threadIdx.x & 63);
  const int r0 = threadIdx.x >> 6;  // 0..3
  float s = 0.f, sq = 0.f;
  const float* p = Hin + (size_t)r0 * H_DIM + col;
  for (int r = r0; r < B_DIM; r += 4) {
    float v = *p;
    p += 4 * (size_t)H_DIM;
    s += v;
    sq += v * v;
  }
  s_sum[threadIdx.x] = s;
  s_sq[threadIdx.x] = sq;
  __syncthreads();
  if (threadIdx.x < 64) {
    const int t = threadIdx.x;
    const float ts = s_sum[t] + s_sum[t + 64] + s_sum[t + 128] + s_sum[t + 192];
    const float tq = s_sq[t] + s_sq[t + 64] + s_sq[t + 128] + s_sq[t + 192];
    const int cc = blockIdx.x * 64 + t;
    const float mu = ts * (1.0f / (float)B_DIM);
    const float var = tq * (1.0f / (float)B_DIM) - mu * mu;
    const float sc = gamma[cc] * rsqrtf(var + BN_EPS);
    aS[cc] = sc;
    bS[cc] = beta[cc] - mu * sc;
  }
}

// ---------------------------------------------------------------------------
// q = relu(aS*h + bS) @ W2 + b2.  h:[B,H], W2T:[E,H], q:[B,E].
// aS/bS staged in LDS (32KB of 320KB WGP pool); 8 waves = 4(M) x 2(N).
// ---------------------------------------------------------------------------
__global__ __launch_bounds__(256) void gemm2_kernel(
    const float* __restrict__ Hin, const float* __restrict__ aS,
    const float* __restrict__ bS, const float* __restrict__ W2T,
    const float* __restrict__ b2, float* __restrict__ Q) {
  __shared__ float s_a[H_DIM];
  __shared__ float s_b[H_DIM];
  for (int i = threadIdx.x; i < H_DIM; i += 256) {
    s_a[i] = aS[i];
    s_b[i] = bS[i];
  }
  __syncthreads();

  const int lane = threadIdx.x & 31;
  const int wave = threadIdx.x >> 5;
  const int lm = lane & 15;
  const int lh = lane >> 4;
  const int row0 = blockIdx.y * 128 + (wave >> 1) * 32;
  const int col0 = blockIdx.x * 64 + (wave & 1) * 32;

  v8f acc00 = {}, acc01 = {}, acc10 = {}, acc11 = {};
  const float* ha0 = Hin + (size_t)(row0 + lm) * H_DIM + 2 * lh;
  const float* ha1 = ha0 + (size_t)16 * H_DIM;
  const float* wb0 = W2T + (size_t)(col0 + lm) * H_DIM + 2 * lh;
  const float* wb1 = wb0 + (size_t)16 * H_DIM;
  for (int k0 = 0; k0 < H_DIM; k0 += 4) {
    const int kk = k0 + 2 * lh;
    v2f sa = *(const v2f*)(&s_a[kk]);
    v2f sb = *(const v2f*)(&s_b[kk]);
    v2f h0 = *(const v2f*)(ha0 + k0);
    v2f h1 = *(const v2f*)(ha1 + k0);
    v2f a0, a1;
    a0.x = fmaxf(h0.x * sa.x + sb.x, 0.0f);
    a0.y = fmaxf(h0.y * sa.y + sb.y, 0.0f);
    a1.x = fmaxf(h1.x * sa.x + sb.x, 0.0f);
    a1.y = fmaxf(h1.y * sa.y + sb.y, 0.0f);
    v2f b0 = *(const v2f*)(wb0 + k0);
    v2f b1 = *(const v2f*)(wb1 + k0);
    acc00 = wmma4(a0, b0, acc00);
    acc01 = wmma4(a0, b1, acc01);
    acc10 = wmma4(a1, b0, acc10);
    acc11 = wmma4(a1, b1, acc11);
  }
  const float bias0 = b2[col0 + lm];
  const float bias1 = b2[col0 + 16 + lm];
  float* q0 = Q + (size_t)(row0 + 8 * lh) * E_DIM + col0 + lm;
  float* q1p = q0 + (size_t)16 * E_DIM;
#pragma unroll
  for (int r = 0; r < 8; ++r) {
    q0[(size_t)r * E_DIM] = acc00[r] + bias0;
    q0[(size_t)r * E_DIM + 16] = acc01[r] + bias1;
    q1p[(size_t)r * E_DIM] = acc10[r] + bias0;
    q1p[(size_t)r * E_DIM + 16] = acc11[r] + bias1;
  }
}

// ---------------------------------------------------------------------------
// invn[row] = 1 / max(||X[row,:]||, 1e-12).  One wave per row (wave32).
// ---------------------------------------------------------------------------
__global__ __launch_bounds__(256) void rownorm_kernel(
    const float* __restrict__ X, float* __restrict__ invn) {
  const int lane = threadIdx.x & 31;
  const int wave = threadIdx.x >> 5;
  const int row = blockIdx.x * 8 + wave;
  const float* p = X + (size_t)row * E_DIM + lane;
  float s = 0.f;
#pragma unroll
  for (int i = 0; i < 8; ++i) {
    float v = p[i * 32];
    s += v * v;
  }
#pragma unroll
  for (int off = 16; off > 0; off >>= 1) s += __shfl_xor(s, off, 32);
  if (lane == 0) invn[row] = 1.0f / fmaxf(sqrtf(s), 1e-12f);
}

// ---------------------------------------------------------------------------
// Normalize + transpose: Out[E,B], Out[e][k] = In[k][e] * invn[k].
// Puts both Gram operands in [E,B] layout -> b64 fragment loads.
// ---------------------------------------------------------------------------
__global__ __launch_bounds__(256) void ntrans_kernel(
    const float* __restrict__ In, const float* __restrict__ invn,
    float* __restrict__ Out) {
  __shared__ float tile[32][33];
  const int lane = threadIdx.x & 31;
  const int grp = threadIdx.x >> 5;
  const int e = blockIdx.x * 32 + lane;
  const int k0 = blockIdx.y * 32 + grp * 4;
#pragma unroll
  for (int i = 0; i < 4; ++i) {
    const int k = k0 + i;
    tile[grp * 4 + i][lane] = In[(size_t)k * E_DIM + e] * invn[k];
  }
  __syncthreads();
  const int ko = blockIdx.y * 32 + lane;
  const int e0 = blockIdx.x * 32 + grp * 4;
#pragma unroll
  for (int i = 0; i < 4; ++i)
    Out[(size_t)(e0 + i) * B_DIM + ko] = tile[lane][grp * 4 + i];
}

// ---------------------------------------------------------------------------
// Gram split-K partials: GP[z] += A_z-tile^ (32x32 wave tiles, 16 K-slices,
// 2 matrices -> grid.z = 32, 256 blocks (2048 waves) of occupancy). No atomics:
// each (mat, slice) writes its own E x E partial; a fixed-order reduction
// kernel makes the result bitwise deterministic.
// A = QnT [E,B] (m index), B = PnT [E,B] (n index), K = batch.
// ---------------------------------------------------------------------------
__global__ __launch_bounds__(256) void gram_kernel(
    const float* __restrict__ A1, const float* __restrict__ Bn1,
    const float* __restrict__ A2, const float* __restrict__ Bn2,
    float* __restrict__ GP) {
  const int mat = blockIdx.z >> 4;
  const int slice = blockIdx.z & 15;
  const float* A = mat ? A2 : A1;
  const float* Bm = mat ? Bn2 : Bn1;

  const int lane = threadIdx.x & 31;
  const int wave = threadIdx.x >> 5;
  const int lm = lane & 15;
  const int lh = lane >> 4;
  const int row0 = blockIdx.y * 64 + (wave >> 2) * 32;
  const int col0 = blockIdx.x * 128 + (wave & 3) * 32;
  const int kb = slice * (B_DIM / 16);

  v8f acc00 = {}, acc01 = {}, acc10 = {}, acc11 = {};
  const float* pa0 = A + (size_t)(row0 + lm) * B_DIM + 2 * lh;
  const float* pa1 = pa0 + (size_t)16 * B_DIM;
  const float* pb0 = Bm + (size_t)(col0 + lm) * B_DIM + 2 * lh;
  const float* pb1 = pb0 + (size_t)16 * B_DIM;
  for (int k0 = kb; k0 < kb + B_DIM / 16; k0 += 4) {
    v2f a0 = *(const v2f*)(pa0 + k0);
    v2f a1 = *(const v2f*)(pa1 + k0);
    v2f b0 = *(const v2f*)(pb0 + k0);
    v2f b1 = *(const v2f*)(pb1 + k0);
    acc00 = wmma4(a0, b0, acc00);
    acc01 = wmma4(a0, b1, acc01);
    acc10 = wmma4(a1, b0, acc10);
    acc11 = wmma4(a1, b1, acc11);
  }
  float* gp0 = GP + (size_t)blockIdx.z * (E_DIM * E_DIM) +
               (size_t)(row0 + 8 * lh) * E_DIM + col0 + lm;
  float* gp1 = gp0 + (size_t)16 * E_DIM;
#pragma unroll
  for (int r = 0; r < 8; ++r) {
    gp0[(size_t)r * E_DIM] = acc00[r];
    gp0[(size_t)r * E_DIM + 16] = acc01[r];
    gp1[(size_t)r * E_DIM] = acc10[r];
    gp1[(size_t)r * E_DIM + 16] = acc11[r];
  }
}

// ---------------------------------------------------------------------------
// Fixed-order split-K reduction + 1/lambda scale; also emits M^T so the M^2
// GEMM and trace contractions read contiguously.
// ---------------------------------------------------------------------------
__global__ __launch_bounds__(256) void gram_reduce_kernel(
    const float* __restrict__ GP, float* __restrict__ M1,
    float* __restrict__ M1T, float* __restrict__ M2, float* __restrict__ M2T,
    const float* __restrict__ lamda_inv) {
  const int idx = blockIdx.x * 256 + threadIdx.x;  // 0 .. 2*E*E-1
  const int mat = idx >> 16;
  const int e = idx & 0xFFFF;
  const float* base = GP + (size_t)mat * 16 * (E_DIM * E_DIM) + e;
  float s = 0.f;
#pragma unroll
  for (int t = 0; t < 16; ++t) s += base[(size_t)t * (E_DIM * E_DIM)];
  s *= 1.0f / lamda_inv[0];
  const int i = e >> 8;
  const int j = e & 255;
  float* M = mat ? M2 : M1;
  float* MT = mat ? M2T : M1T;
  M[e] = s;
  MT[j * E_DIM + i] = s;
}

// ---------------------------------------------------------------------------
// P = M @ M for both 256x256 Gram matrices (A from M row-major, B from M^T);
// also writes P^T for the trace contraction.
// ---------------------------------------------------------------------------
__global__ __launch_bounds__(256) void matsq_kernel(
    const float* __restrict__ M1, const float* __restrict__ M1T,
    const float* __restrict__ M2, const float* __restrict__ M2T,
    float* __restrict__ P1, float* __restrict__ P1T, float* __restrict__ P2,
    float* __restrict__ P2T) {
  const float* M = blockIdx.z ? M2 : M1;
  const float* MT = blockIdx.z ? M2T : M1T;
  float* P = blockIdx.z ? P2 : P1;
  float* PT = blockIdx.z ? P2T : P1T;

  const int lane = threadIdx.x & 31;
  const int wave = threadIdx.x >> 5;
  const int lm = lane & 15;
  const int lh = lane >> 4;
  const int row0 = blockIdx.y * 64 + (wave >> 2) * 32;
  const int col0 = blockIdx.x * 128 + (wave & 3) * 32;

  v8f acc00 = {}, acc01 = {}, acc10 = {}, acc11 = {};
  const float* pa0 = M + (size_t)(row0 + lm) * E_DIM + 2 * lh;
  const float* pa1 = pa0 + (size_t)16 * E_DIM;
  const float* pb0 = MT + (size_t)(col0 + lm) * E_DIM + 2 * lh;
  const float* pb1 = pb0 + (size_t)16 * E_DIM;
  for (int k0 = 0; k0 < E_DIM; k0 += 4) {
    v2f a0 = *(const v2f*)(pa0 + k0);
    v2f a1 = *(const v2f*)(pa1 + k0);
    v2f b0 = *(const v2f*)(pb0 + k0);
    v2f b1 = *(const v2f*)(pb1 + k0);
    acc00 = wmma4(a0, b0, acc00);
    acc01 = wmma4(a0, b1, acc01);
    acc10 = wmma4(a1, b0, acc10);
    acc11 = wmma4(a1, b1, acc11);
  }
#pragma unroll
  for (int r = 0; r < 8; ++r) {
    const int m0r = row0 + 8 * lh + r;
    const int m1r = m0r + 16;
    const int n0 = col0 + lm;
    const int n1 = n0 + 16;
    P[m0r * E_DIM + n0] = acc00[r];
    PT[n0 * E_DIM + m0r] = acc00[r];
    P[m0r * E_DIM + n1] = acc01[r];
    PT[n1 * E_DIM + m0r] = acc01[r];
    P[m1r * E_DIM + n0] = acc10[r];
    PT[n0 * E_DIM + m1r] = acc10[r];
    P[m1r * E_DIM + n1] = acc11[r];
    PT[n1 * E_DIM + m1r] = acc11[r];
  }
}

// ---------------------------------------------------------------------------
// Traces + final scalar, single block, fixed-order (deterministic):
//  t1 = tr M, t2 = <M,M^T>, t3 = <P,M^T>, t4 = <P,P^T>, summed over both mats.
//  out = -lam * 0.5/B * (t1 - t2/2 + t3/3 - t4/4).
// ---------------------------------------------------------------------------
__global__ __launch_bounds__(1024) void trace_final_kernel(
    const float* __restrict__ M1, const float* __restrict__ M1T,
    const float* __restrict__ P1, const float* __restrict__ P1T,
    const float* __restrict__ M2, const float* __restrict__ M2T,
    const float* __restrict__ P2, const float* __restrict__ P2T,
    const float* __restrict__ lamda_inv, float* __restrict__ out) {
  __shared__ float sh[1024];
  const int tid = threadIdx.x;
  float t[4] = {0.f, 0.f, 0.f, 0.f};
  for (int e = tid; e < E_DIM * E_DIM; e += 1024) {
    const int i = e >> 8;
    const int j = e & 255;
    const float m1 = M1[e], mt1 = M1T[e], p1v = P1[e], pt1 = P1T[e];
    const float m2 = M2[e], mt2 = M2T[e], p2v = P2[e], pt2 = P2T[e];
    if (i == j) t[0] += m1 + m2;
    t[1] += m1 * mt1 + m2 * mt2;
    t[2] += p1v * mt1 + p2v * mt2;
    t[3] += p1v * pt1 + p2v * pt2;
  }
  float sums[4] = {0.f, 0.f, 0.f, 0.f};
  for (int v = 0; v < 4; ++v) {
    sh[tid] = t[v];
    __syncthreads();
    for (int s = 512; s > 0; s >>= 1) {
      if (tid < s) sh[tid] += sh[tid + s];
      __syncthreads();
    }
    if (tid == 0) sums[v] = sh[0];
    __syncthreads();
  }
  if (tid == 0) {
    const float S =
        sums[0] - 0.5f * sums[1] + (1.0f / 3.0f) * sums[2] - 0.25f * sums[3];
    const float loss = S * 0.5f / (float)B_DIM;
    out[0] = -loss * lamda_inv[0];
  }
}

// ---------------------------------------------------------------------------
extern "C" void kernel_launch(void* const* d_in, const int* in_sizes, int n_in,
                              void* d_out, int out_size, void* d_ws,
                              size_t ws_size, hipStream_t stream) {
  (void)in_sizes; (void)n_in; (void)out_size; (void)ws_size;
  const float* z1 = (const float*)d_in[0];
  const float* z2 = (const float*)d_in[1];
  const float* p1 = (const float*)d_in[2];
  const float* p2 = (const float*)d_in[3];
  const float* W1 = (const float*)d_in[4];
  const float* gamma = (const float*)d_in[5];
  const float* beta = (const float*)d_in[6];
  const float* W2 = (const float*)d_in[7];
  const float* b2 = (const float*)d_in[8];
  const float* lam = (const float*)d_in[9];
  float* out = (float*)d_out;

  // Workspace layout (floats). h (B*H = 128MB) is dead after the second
  // gemm2, so the normalized transposes and split-K partials overlay it.
  float* ws = (float*)d_ws;
  float* h = ws;                                   // B*H
  float* A1 = h;                                   // q2n^T  [E,B] (overlay)
  float* Bn1 = A1 + (size_t)E_DIM * B_DIM;         // p1n^T  [E,B]
  float* A2 = Bn1 + (size_t)E_DIM * B_DIM;         // q1n^T  [E,B]
  float* Bn2 = A2 + (size_t)E_DIM * B_DIM;         // p2n^T  [E,B]
  float* GP = Bn2 + (size_t)E_DIM * B_DIM;         // 32 * E*E partials
  float* q1 = h + (size_t)B_DIM * H_DIM;           // B*E
  float* q2 = q1 + (size_t)B_DIM * E_DIM;          // B*E
  float* aS = q2 + (size_t)B_DIM * E_DIM;          // H
  float* bS = aS + H_DIM;                          // H
  float* inq1 = bS + H_DIM;                        // B
  float* inq2 = inq1 + B_DIM;                      // B
  float* inp1 = inq2 + B_DIM;                      // B
  float* inp2 = inp1 + B_DIM;                      // B
  float* W1T = inp2 + B_DIM;                       // H*E
  float* W2T = W1T + (size_t)H_DIM * E_DIM;        // E*H
  float* M1 = W2T + (size_t)H_DIM * E_DIM;         // E*E x8 below
  float* M1T = M1 + E_DIM * E_DIM;
  float* M2 = M1T + E_DIM * E_DIM;
  float* M2T = M2 + E_DIM * E_DIM;
  float* P1 = M2T + E_DIM * E_DIM;
  float* P1T = P1 + E_DIM * E_DIM;
  float* P2 = P1T + E_DIM * E_DIM;
  float* P2T = P2 + E_DIM * E_DIM;

  const dim3 blk(256);

  // One-time weight transposes (8MB total traffic).
  transpose_kernel<<<dim3(H_DIM / 32, E_DIM / 32), blk, 0, stream>>>(
      W1, W1T, E_DIM, H_DIM);
  transpose_kernel<<<dim3(E_DIM / 32, H_DIM / 32), blk, 0, stream>>>(
      W2, W2T, H_DIM, E_DIM);

  const dim3 g1(H_DIM / 128, B_DIM / 64);  // gemm1: 32 x 128
  const dim3 g2(E_DIM / 64, B_DIM / 128);  // gemm2: 4 x 64

  // View 1: q1 = predictor(z1)
  gemm1_kernel<<<g1, blk, 0, stream>>>(z1, W1T, h);
  bn_stats_kernel<<<H_DIM / 64, blk, 0, stream>>>(h, gamma, beta, aS, bS);
  gemm2_kernel<<<g2, blk, 0, stream>>>(h, aS, bS, W2T, b2, q1);
  // View 2: q2 = predictor(z2)
  gemm1_kernel<<<g1, blk, 0, stream>>>(z2, W1T, h);
  bn_stats_kernel<<<H_DIM / 64, blk, 0, stream>>>(h, gamma, beta, aS, bS);
  gemm2_kernel<<<g2, blk, 0, stream>>>(h, aS, bS, W2T, b2, q2);

  // Inverse row norms, then normalized transposes into [E,B] layout.
  rownorm_kernel<<<B_DIM / 8, blk, 0, stream>>>(q1, inq1);
  rownorm_kernel<<<B_DIM / 8, blk, 0, stream>>>(q2, inq2);
  rownorm_kernel<<<B_DIM / 8, blk, 0, stream>>>(p1, inp1);
  rownorm_kernel<<<B_DIM / 8, blk, 0, stream>>>(p2, inp2);
  const dim3 gt(E_DIM / 32, B_DIM / 32);
  ntrans_kernel<<<gt, blk, 0, stream>>>(q2, inq2, A1);
  ntrans_kernel<<<gt, blk, 0, stream>>>(p1, inp1, Bn1);
  ntrans_kernel<<<gt, blk, 0, stream>>>(q1, inq1, A2);
  ntrans_kernel<<<gt, blk, 0, stream>>>(p2, inp2, Bn2);

  // Gram via trace cyclicity: M1 = q2n^T p1n / lam, M2 = q1n^T p2n / lam.
  gram_kernel<<<dim3(E_DIM / 128, E_DIM / 64, 32), blk, 0, stream>>>(
      A1, Bn1, A2, Bn2, GP);
  gram_reduce_kernel<<<2 * E_DIM * E_DIM / 256, blk, 0, stream>>>(
      GP, M1, M1T, M2, M2T, lam);
  matsq_kernel<<<dim3(E_DIM / 128, E_DIM / 64, 2), blk, 0, stream>>>(
      M1, M1T, M2, M2T, P1, P1T, P2, P2T);
  trace_final_kernel<<<1, 1024, 0, stream>>>(M1, M1T, P1, P1T, M2, M2T, P2,
                                             P2T, lam, out);
}